// Warp_15685220565630
// MI455X (gfx1250) — compile-verified
//
#include <hip/hip_runtime.h>
#include <hip/hip_bf16.h>

// 3D trilinear warp (grid_sample align_corners=True, zeros padding).
// Memory-bound: ~117 MB compulsory traffic -> ~5us at 23.3 TB/s HBM.
// Image+flow (84 MB) fit in the 192 MB L2, so the 16 gathers/voxel hit L2
// after compulsory fill. No matrix structure -> WMMA not applicable; the
// optimal form is a coalesced wave32 gather kernel with NT output stores.

namespace {

constexpr int DIM  = 128;
constexpr int N3   = DIM * DIM * DIM;   // 2^21 voxels per (batch, plane)
constexpr int BATCH    = 2;
constexpr int CHANNELS = 2;

__global__ __launch_bounds__(256) void warp3d_kernel(
    const float* __restrict__ image,   // [B, C, 128,128,128]
    const float* __restrict__ flow,    // [B, 3, 128,128,128]
    float* __restrict__ out)           // [B, C, 128,128,128]
{
    const int idx     = blockIdx.x * 256 + threadIdx.x;   // [0, B*N3)
    const int spatial = idx & (N3 - 1);
    const int b       = idx >> 21;

    const int w = spatial & (DIM - 1);
    const int h = (spatial >> 7) & (DIM - 1);
    const int d = spatial >> 14;

    // ---- streamed flow read (coalesced along W) -------------------------
    const float* fptr = flow + (size_t)b * 3 * N3 + spatial;
    const float fd = fptr[0];
    const float fh = fptr[N3];
    const float fw = fptr[2 * N3];

    // Speculative prefetch of the next block's flow plane (gfx1250
    // global_prefetch_b8). Stays inside the flow allocation.
    __builtin_prefetch(fptr + 256, 0, 1);

    // ---- trilinear setup ------------------------------------------------
    const float cd = (float)d + fd;
    const float ch = (float)h + fh;
    const float cw = (float)w + fw;

    const float fld = floorf(cd);
    const float flh = floorf(ch);
    const float flw = floorf(cw);
    const float td = cd - fld;
    const float th = ch - flh;
    const float tw = cw - flw;

    const int d0 = (int)fld, h0 = (int)flh, w0 = (int)flw;
    const int d1 = d0 + 1,   h1 = h0 + 1,   w1 = w0 + 1;

    // Validity folded into per-axis weights (zeros padding), indices clamped
    // for the actual gather -- identical semantics to the reference.
    const float wd0 = (1.0f - td) * ((d0 >= 0 && d0 < DIM) ? 1.0f : 0.0f);
    const float wd1 =         td  * ((d1 >= 0 && d1 < DIM) ? 1.0f : 0.0f);
    const float wh0 = (1.0f - th) * ((h0 >= 0 && h0 < DIM) ? 1.0f : 0.0f);
    const float wh1 =         th  * ((h1 >= 0 && h1 < DIM) ? 1.0f : 0.0f);
    const float ww0 = (1.0f - tw) * ((w0 >= 0 && w0 < DIM) ? 1.0f : 0.0f);
    const float ww1 =         tw  * ((w1 >= 0 && w1 < DIM) ? 1.0f : 0.0f);

    const int cd0 = min(max(d0, 0), DIM - 1);
    const int cd1 = min(max(d1, 0), DIM - 1);
    const int ch0 = min(max(h0, 0), DIM - 1);
    const int ch1 = min(max(h1, 0), DIM - 1);
    const int cw0 = min(max(w0, 0), DIM - 1);
    const int cw1 = min(max(w1, 0), DIM - 1);

    // 8 corner weights: bit2->d, bit1->h, bit0->w  (matches reference)
    const float w000 = wd0 * wh0 * ww0;
    const float w001 = wd0 * wh0 * ww1;
    const float w010 = wd0 * wh1 * ww0;
    const float w011 = wd0 * wh1 * ww1;
    const float w100 = wd1 * wh0 * ww0;
    const float w101 = wd1 * wh0 * ww1;
    const float w110 = wd1 * wh1 * ww0;
    const float w111 = wd1 * wh1 * ww1;

    // Row offsets shared by both channels (d*H*W + h*W)
    const int r00 = (cd0 << 14) + (ch0 << 7);
    const int r01 = (cd0 << 14) + (ch1 << 7);
    const int r10 = (cd1 << 14) + (ch0 << 7);
    const int r11 = (cd1 << 14) + (ch1 << 7);

    const float* imgb = image + ((size_t)b * CHANNELS << 21);
    float*       outb = out   + ((size_t)b * CHANNELS << 21);

#pragma unroll
    for (int c = 0; c < CHANNELS; ++c) {
        const float* p = imgb + ((size_t)c << 21);
        float acc;
        acc = p[r00 + cw0] * w000;
        acc = fmaf(p[r00 + cw1], w001, acc);
        acc = fmaf(p[r01 + cw0], w010, acc);
        acc = fmaf(p[r01 + cw1], w011, acc);
        acc = fmaf(p[r10 + cw0], w100, acc);
        acc = fmaf(p[r10 + cw1], w101, acc);
        acc = fmaf(p[r11 + cw0], w110, acc);
        acc = fmaf(p[r11 + cw1], w111, acc);
        // NT store: output is write-once -> don't evict the gather working
        // set (image fits in the 192 MB L2) with the 33.5 MB output stream.
        __builtin_nontemporal_store(acc, outb + ((size_t)c << 21) + spatial);
    }
}

} // namespace

extern "C" void kernel_launch(void* const* d_in, const int* in_sizes, int n_in,
                              void* d_out, int out_size, void* d_ws, size_t ws_size,
                              hipStream_t stream) {
    const float* image = (const float*)d_in[0];  // [2,2,128,128,128] f32
    const float* flow  = (const float*)d_in[1];  // [2,3,128,128,128] f32
    float*       out   = (float*)d_out;          // [2,2,128,128,128] f32
    (void)in_sizes; (void)n_in; (void)out_size; (void)d_ws; (void)ws_size;

    const int total_threads = BATCH * N3;        // 4,194,304 (exact multiple)
    const int block = 256;                       // 8 wave32 waves / block
    const int grid  = total_threads / block;     // 16384 blocks

    hipLaunchKernelGGL(warp3d_kernel, dim3(grid), dim3(block), 0, stream,
                       image, flow, out);
}